// PQLoss_13975823582300
// MI455X (gfx1250) — compile-verified
//
#include <hip/hip_runtime.h>
#include <hip/hip_bf16.h>

typedef float v2f __attribute__((ext_vector_type(2)));
typedef float v8f __attribute__((ext_vector_type(8)));

#define N_PTS 4096
#define D_FEAT 128
#define BIGV 1e10f

// ---------------------------------------------------------------------------
// Kernel 1: row squared norms, nrm[i] = sum_k X[i][k]^2
// ---------------------------------------------------------------------------
__global__ void row_norms_kernel(const float* __restrict__ X,
                                 float* __restrict__ nrm) {
    int i = blockIdx.x * blockDim.x + threadIdx.x;
    if (i >= N_PTS) return;
    const float4* p = (const float4*)(X + (size_t)i * D_FEAT);
    float s = 0.f;
#pragma unroll
    for (int k = 0; k < D_FEAT / 4; ++k) {
        float4 v = p[k];
        s += v.x * v.x + v.y * v.y + v.z * v.z + v.w * v.w;
    }
    nrm[i] = s;
}

// ---------------------------------------------------------------------------
// Kernel 2: D = pairwise distances via fp32 WMMA GEMM (X @ X^T epilogue).
// One wave computes a 32x32 macro-tile as 2x2 of 16x16 WMMA tiles.
// fp32 A 16x4 / B 4x16 fragment layout (wave32):
//   lane = h*16 + m (h = lane>>4, m = lane&15)
//   A: a.x = X[I+m][k+2h], a.y = X[I+m][k+2h+1]
//   B: b.x = X[J+m][k+2h], b.y = X[J+m][k+2h+1]   (B = X^T)
// C/D 16x16 f32: VGPR r, lane h*16+m -> element (row = r + 8h, col = m)
// ---------------------------------------------------------------------------
__global__ void __launch_bounds__(256)
dist_gemm_kernel(const float* __restrict__ X,
                 const float* __restrict__ nrm,
                 float* __restrict__ D) {
    const int lane = threadIdx.x & 31;
    const int wave = threadIdx.x >> 5;
    const int gw   = blockIdx.x * 8 + wave;   // 16384 waves = 128x128 macro-tiles
    const int tI   = (gw >> 7) << 5;
    const int tJ   = (gw & 127) << 5;
    const int h    = lane >> 4;
    const int m    = lane & 15;

    const float* rowA0 = X + (size_t)(tI + m)      * D_FEAT;
    const float* rowA1 = X + (size_t)(tI + 16 + m) * D_FEAT;
    const float* rowB0 = X + (size_t)(tJ + m)      * D_FEAT;
    const float* rowB1 = X + (size_t)(tJ + 16 + m) * D_FEAT;

    v8f c00 = {}; v8f c01 = {}; v8f c10 = {}; v8f c11 = {};

#pragma unroll 4
    for (int k = 0; k < D_FEAT; k += 4) {
        const int ko = k + 2 * h;
        v2f a0 = *(const v2f*)(rowA0 + ko);
        v2f a1 = *(const v2f*)(rowA1 + ko);
        v2f b0 = *(const v2f*)(rowB0 + ko);
        v2f b1 = *(const v2f*)(rowB1 + ko);
        c00 = __builtin_amdgcn_wmma_f32_16x16x4_f32(false, a0, false, b0,
                                                    (short)0, c00, false, false);
        c01 = __builtin_amdgcn_wmma_f32_16x16x4_f32(false, a0, false, b1,
                                                    (short)0, c01, false, false);
        c10 = __builtin_amdgcn_wmma_f32_16x16x4_f32(false, a1, false, b0,
                                                    (short)0, c10, false, false);
        c11 = __builtin_amdgcn_wmma_f32_16x16x4_f32(false, a1, false, b1,
                                                    (short)0, c11, false, false);
    }

    // Epilogue: dot -> distance, diagonal -> BIG
    {
        const v8f* accs[4] = { &c00, &c01, &c10, &c11 };
#pragma unroll
        for (int t = 0; t < 4; ++t) {
            const int bi = t >> 1, bj = t & 1;
            const int col = tJ + bj * 16 + m;
            const float nc = nrm[col];
            const v8f& c = *accs[t];
#pragma unroll
            for (int r = 0; r < 8; ++r) {
                const int row = tI + bi * 16 + 8 * h + r;
                float d2 = nrm[row] + nc - 2.0f * c[r];
                float dist = sqrtf(fmaxf(d2, 0.0f) + 1e-12f);
                if (row == col) dist = BIGV;
                D[(size_t)row * N_PTS + col] = dist;
            }
        }
    }
}

// ---------------------------------------------------------------------------
// Kernel 3: Prim's MST (H0 deaths) + pq loss, single persistent block.
// 1024 threads, each owns 4 points (tid + 1024k). mind[] in registers,
// visited as 4-bit register mask; argmin = wave shuffle + LDS cross-wave,
// tie-break toward lowest global index (matches jnp.argmin).
// ---------------------------------------------------------------------------
__global__ void __launch_bounds__(1024)
prim_loss_kernel(const float* __restrict__ D, float* __restrict__ out) {
    __shared__ float red_val[32];
    __shared__ int   red_idx[32];
    __shared__ float s_death;
    __shared__ int   s_j;

    const int tid  = threadIdx.x;
    const int lane = tid & 31;
    const int wave = tid >> 5;

    float mind[4];
    unsigned vis = 0;
#pragma unroll
    for (int k = 0; k < 4; ++k)
        mind[k] = D[(size_t)(tid + (k << 10))];   // row 0 of D
    if (tid == 0) vis |= 1u;                      // point 0 starts visited

    double acc = 0.0;

    for (int it = 0; it < N_PTS - 1; ++it) {
        // local argmin over owned 4 entries (visited masked to BIG)
        float best = BIGV * 2.0f;
        int   bidx = N_PTS;
#pragma unroll
        for (int k = 0; k < 4; ++k) {
            const int i = tid + (k << 10);
            const float v = ((vis >> k) & 1u) ? BIGV : mind[k];
            if (v < best || (v == best && i < bidx)) { best = v; bidx = i; }
        }
        // wave32 reduce
#pragma unroll
        for (int off = 16; off > 0; off >>= 1) {
            const float ov = __shfl_xor(best, off, 32);
            const int   oi = __shfl_xor(bidx, off, 32);
            if (ov < best || (ov == best && oi < bidx)) { best = ov; bidx = oi; }
        }
        if (lane == 0) { red_val[wave] = best; red_idx[wave] = bidx; }
        __syncthreads();
        if (wave == 0) {
            best = red_val[lane];
            bidx = red_idx[lane];
#pragma unroll
            for (int off = 16; off > 0; off >>= 1) {
                const float ov = __shfl_xor(best, off, 32);
                const int   oi = __shfl_xor(bidx, off, 32);
                if (ov < best || (ov == best && oi < bidx)) { best = ov; bidx = oi; }
            }
            if (lane == 0) { s_death = best; s_j = bidx; }
        }
        __syncthreads();
        const int   j     = s_j;
        const float death = s_death;

        if (tid == 0) {
            // pers^p * mid^q with births=0, p=2, q=1  ->  0.5 * d^3
            const double d = (double)death;
            acc += 0.5 * d * d * d;
        }
        if (tid == (j & 1023)) vis |= (1u << (j >> 10));

        const float* row = D + (size_t)j * N_PTS;
#pragma unroll
        for (int k = 0; k < 4; ++k) {
            const int i = tid + (k << 10);
            mind[k] = fminf(mind[k], row[i]);   // unmasked, same as reference
        }
    }

    if (tid == 0) out[0] = (float)acc;
}

// ---------------------------------------------------------------------------
extern "C" void kernel_launch(void* const* d_in, const int* in_sizes, int n_in,
                              void* d_out, int out_size, void* d_ws, size_t ws_size,
                              hipStream_t stream) {
    (void)in_sizes; (void)n_in; (void)out_size; (void)ws_size;

    const float* X = (const float*)d_in[0];
    float* out = (float*)d_out;

    float* D   = (float*)d_ws;                                       // 64 MB
    float* nrm = (float*)((char*)d_ws +
                          (size_t)N_PTS * N_PTS * sizeof(float));    // +16 KB

    row_norms_kernel<<<N_PTS / 256, 256, 0, stream>>>(X, nrm);
    dist_gemm_kernel<<<(128 * 128) / 8, 256, 0, stream>>>(X, nrm, D);
    prim_loss_kernel<<<1, 1024, 0, stream>>>(D, out);
}